// Model_51788715655811
// MI455X (gfx1250) — compile-verified
//
#include <hip/hip_runtime.h>
#include <hip/hip_bf16.h>
#include <math.h>

#define BB   16
#define NN   5000
#define FIN  32
#define DD   64
#define EE   80000
#define NEDG (EE + NN)       // edges + self loops = 85000
#define OBSK (NN * DD)       // 320000
#define KTILE 64             // K-tile for LDS-staged head GEMM

typedef __attribute__((ext_vector_type(2))) float v2f;
typedef __attribute__((ext_vector_type(8))) float v8f;
typedef int v4i_ __attribute__((vector_size(4 * sizeof(int))));

#define AS1 __attribute__((address_space(1)))
#define AS3 __attribute__((address_space(3)))

// D = A(16x4 f32) * B(4x16 f32) + C(16x16 f32)
__device__ __forceinline__ v8f wmma4(v2f a, v2f b, v8f c) {
  return __builtin_amdgcn_wmma_f32_16x16x4_f32(
      false, a, false, b, (short)0, c, false, false);
}

// ---- CDNA5 async global->LDS copy (16 bytes / lane), with portable fallback
#if __has_builtin(__builtin_amdgcn_global_load_async_to_lds_b128)
#define HAVE_ASYNC 1
__device__ __forceinline__ void async_cp16(const float* g, float* l) {
  __builtin_amdgcn_global_load_async_to_lds_b128(
      (AS1 v4i_*)g, (AS3 v4i_*)l, 0, 0);
}
#else
#define HAVE_ASYNC 0
__device__ __forceinline__ void async_cp16(const float* g, float* l) {
  float4 t = *(const float4*)g;       // global_load_b128
  *(float4*)l = t;                    // ds_store_b128
}
#endif

// ---------------------------------------------------------------- utilities
__global__ void k_zero(float* p, long n) {
  long i = (long)blockIdx.x * blockDim.x + threadIdx.x;
  long stride = (long)gridDim.x * blockDim.x;
  for (; i < n; i += stride) p[i] = 0.0f;
}

// ------------------------------------------------- CSR build (by destination)
__global__ void k_cnt_init(int* __restrict__ cnt, int* __restrict__ cur) {
  int i = blockIdx.x * blockDim.x + threadIdx.x;
  if (i < NN) { cnt[i] = 1; cur[i] = 0; }       // self-loop counts 1
}

__global__ void k_cnt_accum(const int* __restrict__ dst, int* __restrict__ cnt) {
  int e = blockIdx.x * blockDim.x + threadIdx.x;
  if (e < EE) atomicAdd(&cnt[dst[e]], 1);
}

__global__ void k_scan(const int* __restrict__ cnt, int* __restrict__ rs) {
  if (threadIdx.x == 0 && blockIdx.x == 0) {
    int acc = 0;
    for (int i = 0; i < NN; ++i) { rs[i] = acc; acc += cnt[i]; }
    rs[NN] = acc;
  }
}

__global__ void k_dinv(const int* __restrict__ cnt, float* __restrict__ dinv) {
  int i = blockIdx.x * blockDim.x + threadIdx.x;
  if (i < NN) dinv[i] = rsqrtf((float)cnt[i]);
}

__global__ void k_fill(const int* __restrict__ ei, const int* __restrict__ rs,
                       int* __restrict__ cur, int* __restrict__ srcl) {
  int e = blockIdx.x * blockDim.x + threadIdx.x;
  if (e >= NEDG) return;
  int s, t;
  if (e < EE) { s = ei[e]; t = ei[EE + e]; }
  else        { s = e - EE; t = s; }            // self loop
  int pos = atomicAdd(&cur[t], 1);
  srcl[rs[t] + pos] = s;
}

// ------------------------------------------------- GCN feature-transform GEMM
// Y[rows, Dout] = X[rows, K] @ W[K, Dout]; one wave per 16x16 tile
__global__ void k_gemm_rows(const float* __restrict__ X, const float* __restrict__ W,
                            float* __restrict__ Y, int K, int Dout) {
  const int lane = threadIdx.x;
  const int half = lane >> 4;
  const int l16  = lane & 15;
  const int r0 = blockIdx.x * 16;
  const int c0 = blockIdx.y * 16;
  const float* xrow = X + (long)(r0 + l16) * K;
  v8f acc = {};
  for (int k = 0; k < K; k += 4) {
    v2f a, b;
    a.x = xrow[k + 2 * half];
    a.y = xrow[k + 2 * half + 1];
    b.x = W[(long)(k + 2 * half)     * Dout + c0 + l16];
    b.y = W[(long)(k + 2 * half + 1) * Dout + c0 + l16];
    acc = wmma4(a, b, acc);
  }
#pragma unroll
  for (int i = 0; i < 8; ++i)
    Y[(long)(r0 + i + 8 * half) * Dout + c0 + l16] = acc[i];
}

// ----------------- pull-style GCN aggregation, fused bias+tanh, atomic-free
// Y[b,n,:] = tanh( dinv[n] * sum_{s in nbr(n)} dinv[s]*h[b,s,:] + bias )
__global__ void k_gather_tanh(const float* __restrict__ h, const int* __restrict__ rs,
                              const int* __restrict__ srcl, const float* __restrict__ dinv,
                              const float* __restrict__ bias, float* __restrict__ Y) {
  int idx = blockIdx.x * blockDim.x + threadIdx.x;   // BB*NN*16 quads
  if (idx >= BB * NN * 16) return;
  int q = idx & 15;
  int rest = idx >> 4;
  int n = rest % NN;
  int b = rest / NN;
  int d0 = q * 4;
  const int j0 = rs[n], j1 = rs[n + 1];
  const float* hb = h + (long)b * NN * DD + d0;
  float a0 = 0.f, a1 = 0.f, a2 = 0.f, a3 = 0.f;
  for (int j = j0; j < j1; ++j) {
    int s = srcl[j];
    float w = dinv[s];
    const float4 v = *(const float4*)(hb + (long)s * DD);
    a0 += v.x * w; a1 += v.y * w; a2 += v.z * w; a3 += v.w * w;
  }
  float dn = dinv[n];
  long o = ((long)b * NN + n) * DD + d0;
  Y[o + 0] = tanhf(a0 * dn + bias[d0 + 0]);
  Y[o + 1] = tanhf(a1 * dn + bias[d0 + 1]);
  Y[o + 2] = tanhf(a2 * dn + bias[d0 + 2]);
  Y[o + 3] = tanhf(a3 * dn + bias[d0 + 3]);
}

__global__ void k_bias_tanh(const float* __restrict__ X, const float* __restrict__ bias,
                            float* __restrict__ Y, long n, int Dc) {
  long i = (long)blockIdx.x * blockDim.x + threadIdx.x;
  long stride = (long)gridDim.x * blockDim.x;
  for (; i < n; i += stride) Y[i] = tanhf(X[i] + bias[(int)(i % Dc)]);
}

// --------- head layer-1: [16,Ktot]@[Ktot,Dout], K-split, async-LDS pipelined
__global__ void k_gemm16_lds(const float* __restrict__ X, const float* __restrict__ W,
                             float* __restrict__ Y, int Ktot, int Dout, int kchunk) {
  __shared__ __align__(16) float sA[2][16 * 68];     // padded: 68 dwords/row
  __shared__ __align__(16) float sB[2][KTILE * 16];
  const int lane = threadIdx.x, half = lane >> 4, l16 = lane & 15;
  const int c0 = blockIdx.x * 16;
  const int kbeg = blockIdx.y * kchunk;
  const int T = kchunk / KTILE;

  auto issue = [&](int t, int nb) {
    const int k0 = kbeg + t * KTILE;
    const float* wb = W + (long)k0 * Dout + c0;
    const float* ab = X + k0;
#pragma unroll
    for (int s = 0; s < 8; ++s) {                    // B: 64 rows x 16 cols
      int f = s * 32 + lane;                         // 0..255 b128 groups
      int row = f >> 2, grp = f & 3;
      async_cp16(wb + (long)row * Dout + grp * 4, &sB[nb][f * 4]);
    }
#pragma unroll
    for (int s = 0; s < 8; ++s) {                    // A: 16 rows x 64 k
      int f = s * 32 + lane;
      int row = f >> 4, grp = f & 15;
      async_cp16(ab + (long)row * Ktot + grp * 4, &sA[nb][row * 68 + grp * 4]);
    }
  };

  v8f acc = {};
  issue(0, 0);
  for (int t = 0; t < T; ++t) {
    const int nb = t & 1;
    if (t + 1 < T) {
      asm volatile("s_wait_dscnt 0x0" ::: "memory");   // prev reads of buf nb^1 done
      issue(t + 1, nb ^ 1);
#if HAVE_ASYNC
      asm volatile("s_wait_asynccnt 0x10" ::: "memory"); // tile t (16 issues) complete
#endif
    } else {
#if HAVE_ASYNC
      asm volatile("s_wait_asynccnt 0x0" ::: "memory");
#endif
    }
#pragma unroll
    for (int kk = 0; kk < KTILE / 4; ++kk) {
      int kr = kk * 4 + 2 * half;
      v2f a, b;
      a.x = sA[nb][l16 * 68 + kr];
      a.y = sA[nb][l16 * 68 + kr + 1];
      b.x = sB[nb][kr * 16 + l16];
      b.y = sB[nb][(kr + 1) * 16 + l16];
      acc = wmma4(a, b, acc);
    }
  }
#pragma unroll
  for (int i = 0; i < 8; ++i)
    unsafeAtomicAdd(&Y[(long)(i + 8 * half) * Dout + c0 + l16], acc[i]);
}

// --------------------------- head layers 2/3: [16,K]@[K,Dout], full K per wave
__global__ void k_gemm16_full(const float* __restrict__ X, const float* __restrict__ W,
                              float* __restrict__ Y, int K, int Dout) {
  const int lane = threadIdx.x, half = lane >> 4, l16 = lane & 15;
  const int col = blockIdx.x * 16 + l16;
  const bool inb = col < Dout;
  const float* xrow = X + (long)l16 * K;
  v8f acc = {};
  for (int k = 0; k < K; k += 4) {
    v2f a, b;
    a.x = xrow[k + 2 * half];
    a.y = xrow[k + 2 * half + 1];
    b.x = inb ? W[(long)(k + 2 * half)     * Dout + col] : 0.0f;
    b.y = inb ? W[(long)(k + 2 * half + 1) * Dout + col] : 0.0f;
    acc = wmma4(a, b, acc);
  }
  if (inb) {
#pragma unroll
    for (int i = 0; i < 8; ++i) Y[(long)(i + 8 * half) * Dout + col] = acc[i];
  }
}

__global__ void k_critic_l3(const float* __restrict__ hc2, const float* __restrict__ Wc3,
                            const float* __restrict__ bc3, float* __restrict__ val) {
  int b = threadIdx.x;
  if (b < BB) {
    float s = 0.0f;
    for (int k = 0; k < 64; ++k) s += hc2[b * 64 + k] * Wc3[k];
    val[b] = s + bc3[0];
  }
}

// ----------------------------------------------------------- Beta-head math
__device__ __forceinline__ float digammaf_(float x) {
  float r = 0.0f;
  while (x < 6.0f) { r -= 1.0f / x; x += 1.0f; }
  float inv = 1.0f / x, i2 = inv * inv;
  return r + logf(x) - 0.5f * inv
         - i2 * (1.0f / 12.0f - i2 * (1.0f / 120.0f - i2 * (1.0f / 252.0f)));
}

__device__ __forceinline__ float softplus_(float x) {
  return (x > 20.0f) ? x : log1pf(expf(x));
}

__global__ void k_head(const float* __restrict__ ha3, const float* __restrict__ ba3,
                       const float* __restrict__ hb3, const float* __restrict__ bb3,
                       float* __restrict__ out) {
  int idx = blockIdx.x * blockDim.x + threadIdx.x;
  if (idx >= BB * NN) return;
  int n = idx % NN;
  float alpha = 1.0f + softplus_(ha3[idx] + ba3[n]);
  float beta  = 1.0f + softplus_(hb3[idx] + bb3[n]);
  float action = alpha / (alpha + beta);
  float logB = lgammaf(alpha) + lgammaf(beta) - lgammaf(alpha + beta);
  float logp = (alpha - 1.0f) * logf(action) + (beta - 1.0f) * log1pf(-action) - logB;
  float ent  = logB - (alpha - 1.0f) * digammaf_(alpha)
                    - (beta  - 1.0f) * digammaf_(beta)
                    + (alpha + beta - 2.0f) * digammaf_(alpha + beta);
  out[idx]               = action;
  out[BB * NN + idx]     = logp;
  out[2 * BB * NN + idx] = ent;
}

// ------------------------------------------------------------------- launch
extern "C" void kernel_launch(void* const* d_in, const int* in_sizes, int n_in,
                              void* d_out, int out_size, void* d_ws, size_t ws_size,
                              hipStream_t stream) {
  (void)in_sizes; (void)n_in; (void)out_size; (void)ws_size;
  const float* x   = (const float*)d_in[0];
  const int*   ei  = (const int*)  d_in[1];
  const float* W1  = (const float*)d_in[2];  const float* b1  = (const float*)d_in[3];
  const float* W2  = (const float*)d_in[4];  const float* b2  = (const float*)d_in[5];
  const float* Wc1 = (const float*)d_in[6];  const float* bc1 = (const float*)d_in[7];
  const float* Wc2 = (const float*)d_in[8];  const float* bc2 = (const float*)d_in[9];
  const float* Wc3 = (const float*)d_in[10]; const float* bc3 = (const float*)d_in[11];
  const float* Wa1 = (const float*)d_in[12]; const float* ba1 = (const float*)d_in[13];
  const float* Wa2 = (const float*)d_in[14]; const float* ba2 = (const float*)d_in[15];
  const float* Wa3 = (const float*)d_in[16]; const float* ba3 = (const float*)d_in[17];
  const float* Wb1 = (const float*)d_in[18]; const float* bb1 = (const float*)d_in[19];
  const float* Wb2 = (const float*)d_in[20]; const float* bb2 = (const float*)d_in[21];
  const float* Wb3 = (const float*)d_in[22]; const float* bb3 = (const float*)d_in[23];
  float* out = (float*)d_out;

  float* fws  = (float*)d_ws;
  float* dinv = fws;  fws += 8192;
  float* bufA = fws;  fws += (long)BB * NN * DD;   // 5.12M floats
  float* bufB = fws;  fws += (long)BB * NN * DD;   // 5.12M floats
  float* hc1  = fws;  fws += 1024;
  float* hc2  = fws;  fws += 1024;
  float* ha1  = fws;  fws += 2048;
  float* ha2  = fws;  fws += 2048;
  float* ha3  = fws;  fws += BB * NN;
  float* hb1  = fws;  fws += 2048;
  float* hb2  = fws;  fws += 2048;
  float* hb3  = fws;  fws += BB * NN;
  int* iws  = (int*)fws;
  int* cnt  = iws;  iws += 8192;
  int* rs   = iws;  iws += 8192;      // rowstart[NN+1]
  int* cur  = iws;  iws += 8192;
  int* srcl = iws;  iws += 90112;     // NEDG

  // CSR by destination (built once, reused by both GCN layers)
  k_cnt_init<<<(NN + 255) / 256, 256, 0, stream>>>(cnt, cur);
  k_cnt_accum<<<(EE + 255) / 256, 256, 0, stream>>>(ei + EE, cnt);
  k_scan<<<1, 1, 0, stream>>>(cnt, rs);
  k_dinv<<<(NN + 255) / 256, 256, 0, stream>>>(cnt, dinv);
  k_fill<<<(NEDG + 255) / 256, 256, 0, stream>>>(ei, rs, cur, srcl);

  const unsigned gatherBlocks = (BB * NN * 16 + 255) / 256;

  // GCN layer 1: h = x@W1 -> pull-gather + bias + tanh (no atomics, no zeroing)
  k_gemm_rows<<<dim3(BB * NN / 16, DD / 16), 32, 0, stream>>>(x, W1, bufA, FIN, DD);
  k_gather_tanh<<<gatherBlocks, 256, 0, stream>>>(bufA, rs, srcl, dinv, b1, bufB);

  // GCN layer 2
  k_gemm_rows<<<dim3(BB * NN / 16, DD / 16), 32, 0, stream>>>(bufB, W2, bufA, DD, DD);
  k_gather_tanh<<<gatherBlocks, 256, 0, stream>>>(bufA, rs, srcl, dinv, b2, bufB);
  // bufB now holds obs [16, 320000]

  // zero layer-1 head accumulators
  k_zero<<<8, 256, 0, stream>>>(hc1, 1024);
  k_zero<<<8, 256, 0, stream>>>(ha1, 2048);
  k_zero<<<8, 256, 0, stream>>>(hb1, 2048);

  // layer-1 heads: 410 MB weight stream, async-LDS double-buffered WMMA
  const int KCH = 1600, NCH = OBSK / KCH;  // 200 chunks x 25 K-tiles of 64
  k_gemm16_lds<<<dim3(64 / 16, NCH), 32, 0, stream>>>(bufB, Wc1, hc1, OBSK, 64, KCH);
  k_gemm16_lds<<<dim3(128 / 16, NCH), 32, 0, stream>>>(bufB, Wa1, ha1, OBSK, 128, KCH);
  k_gemm16_lds<<<dim3(128 / 16, NCH), 32, 0, stream>>>(bufB, Wb1, hb1, OBSK, 128, KCH);

  // critic tail
  k_bias_tanh<<<4, 256, 0, stream>>>(hc1, bc1, hc1, 1024, 64);
  k_gemm16_full<<<4, 32, 0, stream>>>(hc1, Wc2, hc2, 64, 64);
  k_bias_tanh<<<4, 256, 0, stream>>>(hc2, bc2, hc2, 1024, 64);
  k_critic_l3<<<1, 32, 0, stream>>>(hc2, Wc3, bc3, out + 3 * BB * NN);

  // actor alpha tail
  k_bias_tanh<<<8, 256, 0, stream>>>(ha1, ba1, ha1, 2048, 128);
  k_gemm16_full<<<8, 32, 0, stream>>>(ha1, Wa2, ha2, 128, 128);
  k_bias_tanh<<<8, 256, 0, stream>>>(ha2, ba2, ha2, 2048, 128);
  k_gemm16_full<<<(NN + 15) / 16, 32, 0, stream>>>(ha2, Wa3, ha3, 128, NN);

  // actor beta tail
  k_bias_tanh<<<8, 256, 0, stream>>>(hb1, bb1, hb1, 2048, 128);
  k_gemm16_full<<<8, 32, 0, stream>>>(hb1, Wb2, hb2, 128, 128);
  k_bias_tanh<<<8, 256, 0, stream>>>(hb2, bb2, hb2, 2048, 128);
  k_gemm16_full<<<(NN + 15) / 16, 32, 0, stream>>>(hb2, Wb3, hb3, 128, NN);

  // Beta distribution outputs: action, logp, entropy
  k_head<<<(BB * NN + 255) / 256, 256, 0, stream>>>(ha3, ba3, hb3, bb3, out);
}